// GraphConvolution_80702435492057
// MI455X (gfx1250) — compile-verified
//
#include <hip/hip_runtime.h>
#include <hip/hip_bf16.h>
#include <cstdint>

#define NS 1024
#define NT 256
#define NA 256
#define DH 64
#define MASK_VAL -11111.0f

typedef __attribute__((ext_vector_type(2))) float v2f;
typedef __attribute__((ext_vector_type(4))) float v4f;
typedef __attribute__((ext_vector_type(8))) float v8f;

// gfx1250 async global->LDS path (guarded: falls back to float4 copies)
#if __has_builtin(__builtin_amdgcn_global_load_async_to_lds_b128) && \
    __has_builtin(__builtin_amdgcn_s_wait_asynccnt)
#define ASYNC_OK 1
// builtin signature (from compiler diagnostic): (int4 AS1*, int4 AS3*, Ii, Ii)
typedef __attribute__((ext_vector_type(4))) int i4;
typedef __attribute__((address_space(1))) i4 gi4;   // int4 in global memory
typedef __attribute__((address_space(3))) i4 li4;   // int4 in LDS
#define ASG(p) ((gi4*)(uintptr_t)(p))
#define ASL(p) ((li4*)(uintptr_t)(p))
#else
#define ASYNC_OK 0
#endif

// ---------------------------------------------------------------------------
// Kernel 1: attribute transform GEMM via f32 WMMA.
//   out[n,d] = sum_k feat[n,k] * W[d,k] + bias[d]
// One 256-thread block = 8 wave32s; each wave computes a 16(node) x 16(d)
// f32 tile with 16 chained V_WMMA_F32_16X16X4_F32 (K = 64).
// Grid sized exactly -> EXEC is all-ones for every wave (WMMA requirement).
// ---------------------------------------------------------------------------
__global__ void attr_gemm_wmma(const float* __restrict__ feat,
                               const float* __restrict__ W,
                               const float* __restrict__ bias,
                               float* __restrict__ out) {
    const int wave = threadIdx.x >> 5;        // 0..7
    const int lane = threadIdx.x & 31;
    const int mtile = blockIdx.x * 32 + (wave >> 2) * 16;  // node tile
    const int dtile = (wave & 3) * 16;                     // output-dim tile
    const int lm    = lane & 15;
    const int khalf = (lane >> 4) * 2;        // 0 for lanes 0-15, 2 for 16-31

    // A (16x4, MxK): lanes 0-15 hold K={k0,k0+1} of row M=lane,
    //                lanes 16-31 hold K={k0+2,k0+3} of row M=lane-16.
    // B (4x16, KxN): same striping with N = lane&15, B[k,n] = W[n,k].
    const float* __restrict__ arow = feat + (mtile + lm) * DH;
    const float* __restrict__ brow = W    + (dtile + lm) * DH;

    v8f c = {};
#pragma unroll
    for (int k0 = 0; k0 < DH; k0 += 4) {
        v2f a, b;
        a.x = arow[k0 + khalf];
        a.y = arow[k0 + khalf + 1];
        b.x = brow[k0 + khalf];
        b.y = brow[k0 + khalf + 1];
        c = __builtin_amdgcn_wmma_f32_16x16x4_f32(
                /*neg_a=*/false, a, /*neg_b=*/false, b,
                /*c_mod=*/(short)0, c, /*reuse_a=*/false, /*reuse_b=*/false);
    }

    // C/D layout: VGPR r -> lanes 0-15: M=r, N=lane; lanes 16-31: M=r+8.
    const int row0 = mtile + ((lane >> 4) << 3);
    const int col  = dtile + lm;
    const float bv = bias[col];
#pragma unroll
    for (int r = 0; r < 8; ++r)
        out[(row0 + r) * DH + col] = c[r] + bv;
}

// ---------------------------------------------------------------------------
// Kernel 2: 14 alpha projections  proj[k][n] = feat_k[n,:] . alpha_W[k] + b[k]
// ---------------------------------------------------------------------------
__global__ void proj_kernel(const float* __restrict__ sf,
                            const float* __restrict__ tf,
                            const float* __restrict__ af,
                            const float* __restrict__ alpha_W,
                            const float* __restrict__ alpha_b,
                            float* __restrict__ proj) {
    const int k = blockIdx.y;                 // 0..13
    const int node = blockIdx.x * 128 + threadIdx.x;
    const int fsel[14] = {0,0, 0,1, 1,0, 1,1, 0,2, 2,0, 2,2};
    const int f = fsel[k];
    const int n = (f == 0) ? NS : ((f == 1) ? NT : NA);
    if (node >= n) return;
    const float* feat = (f == 0) ? sf : ((f == 1) ? tf : af);
    const float* w = alpha_W + k * DH;
    float acc = alpha_b[k];
#pragma unroll 8
    for (int d = 0; d < DH; ++d) acc += feat[node * DH + d] * w[d];
    proj[k * NS + node] = acc;
}

// ---------------------------------------------------------------------------
// Kernel 3: per-row softmax statistics (max, sum-of-exp) for one attention
// pair. Masked entries use MASK_VAL; exp(MASK-m) underflows to 0 exactly
// whenever the row has a live entry (elu >= -1), matching the reference.
// ---------------------------------------------------------------------------
__global__ void rowstats_kernel(const float* __restrict__ pa,
                                const float* __restrict__ pb,
                                const int* __restrict__ adj, int adj_stride,
                                int nj,
                                float* __restrict__ m_out,
                                float* __restrict__ z_out) {
    const int i = blockIdx.x;
    const int tid = threadIdx.x;              // 256 threads
    const float pai = pa[i];

    float s_local[4];                          // nj <= 1024 -> <=4 per thread
    int cnt = 0;
    float mloc = -3.0e38f;
    for (int j = tid; j < nj; j += 256) {
        float s;
        if (adj[i * adj_stride + j] != 0) {
            float x = pai + pb[j];
            s = (x > 0.f) ? x : (__expf(x) - 1.f);   // elu
        } else {
            s = MASK_VAL;
        }
        s_local[cnt++] = s;
        mloc = fmaxf(mloc, s);
    }

    __shared__ float red[256];
    red[tid] = mloc;
    __syncthreads();
    for (int off = 128; off > 0; off >>= 1) {
        if (tid < off) red[tid] = fmaxf(red[tid], red[tid + off]);
        __syncthreads();
    }
    const float m = red[0];
    __syncthreads();

    float zloc = 0.f;
    for (int c = 0; c < cnt; ++c) zloc += __expf(s_local[c] - m);
    red[tid] = zloc;
    __syncthreads();
    for (int off = 128; off > 0; off >>= 1) {
        if (tid < off) red[tid] += red[tid + off];
        __syncthreads();
    }
    if (tid == 0) { m_out[i] = m; z_out[i] = red[0]; }
}

// ---------------------------------------------------------------------------
// Kernel 4: fused gated aggregation with double-buffered LDS tiles.
//   out[i,d] = residual[i,d] + sum_seg sum_j w[i,j]*elu(ta[i,d]+gb[j,d])*feat[j,d]
// Block = 256 threads = 4 output rows x 64 dims. j-tiles of 32 staged into
// LDS with GLOBAL_LOAD_ASYNC_TO_LDS_B128 (ASYNCcnt) when available; the next
// tile's loads are in flight while the current tile's elu/FMA loop runs.
// ---------------------------------------------------------------------------
struct Seg {
    const float* gb;    // t_x of the j side        (nj x 64)
    const float* feat;  // raw feature of the j side (nj x 64)
    const float* pb;    // alpha projection of j side
    const float* pa;    // alpha projection of i side
    const float* m;     // per-row softmax max
    const float* z;     // per-row softmax denom
    const int*   adj;   // adjacency base at [i=0, j=0] of this sub-block
    int adj_stride;
    int nj;
};
struct SegPack { Seg s[3]; int nseg; };

#define TJ 32           // j-tile; TJ*DH = 2048 floats = 512 float4 per array

__device__ __forceinline__ void issue_tile(const Seg& sg, int jt,
                                           float* bgb, float* bft, int tid) {
    const float* g1 = sg.gb   + jt * DH;
    const float* g2 = sg.feat + jt * DH;
#if ASYNC_OK
    const int b0 = tid * 16;            // byte offset of float4 #tid
    const int b1 = (tid + 256) * 16;
    __builtin_amdgcn_global_load_async_to_lds_b128(
        ASG((const char*)g1 + b0), ASL((char*)bgb + b0), 0, 0);
    __builtin_amdgcn_global_load_async_to_lds_b128(
        ASG((const char*)g1 + b1), ASL((char*)bgb + b1), 0, 0);
    __builtin_amdgcn_global_load_async_to_lds_b128(
        ASG((const char*)g2 + b0), ASL((char*)bft + b0), 0, 0);
    __builtin_amdgcn_global_load_async_to_lds_b128(
        ASG((const char*)g2 + b1), ASL((char*)bft + b1), 0, 0);
#else
    v4f*       d1 = (v4f*)bgb;
    v4f*       d2 = (v4f*)bft;
    const v4f* s1 = (const v4f*)g1;
    const v4f* s2 = (const v4f*)g2;
    d1[tid] = s1[tid];  d1[tid + 256] = s1[tid + 256];
    d2[tid] = s2[tid];  d2[tid + 256] = s2[tid + 256];
#endif
}

__device__ __forceinline__ void wait_tiles() {
#if ASYNC_OK
    __builtin_amdgcn_s_wait_asynccnt(0);
#endif
}

__global__ void agg_kernel(SegPack pack,
                           const float* __restrict__ ta,
                           const float* __restrict__ residual,
                           float* __restrict__ out) {
    __shared__ __align__(16) float sh_gb[2][TJ * DH];
    __shared__ __align__(16) float sh_ft[2][TJ * DH];
    __shared__ float sh_w[2][4 * TJ];

    const int tid = threadIdx.x;              // 256
    const int il  = tid >> 6;                 // local row 0..3
    const int d   = tid & 63;
    const int i   = blockIdx.x * 4 + il;

    const float ga = ta[i * DH + d];
    float acc = residual ? residual[i * DH + d] : 0.f;

    // flatten all segments' tiles into one sequence so the double-buffer
    // reuse invariant holds across segment boundaries
    int ntiles[3]; int total = 0;
    for (int s = 0; s < 3; ++s) {
        ntiles[s] = (s < pack.nseg) ? (pack.s[s].nj / TJ) : 0;
        total += ntiles[s];
    }
    auto tile_seg = [&](int g, int& jt) -> int {
        for (int s = 0; s < 3; ++s) {
            if (g < ntiles[s]) { jt = g * TJ; return s; }
            g -= ntiles[s];
        }
        jt = 0; return 0;
    };

    // prologue: tile 0 -> buffer 0
    { int jt0; int s0 = tile_seg(0, jt0);
      issue_tile(pack.s[s0], jt0, &sh_gb[0][0], &sh_ft[0][0], tid); }

    for (int g = 0; g < total; ++g) {
        const int p = g & 1;
        int jt; const int cs = tile_seg(g, jt);
        const Seg& sg = pack.s[cs];

        wait_tiles();        // tile g resident (this wave's async loads)
        __syncthreads();     // all waves loaded; buf p^1 readers (g-1) retired

        if (g + 1 < total) { // overlap: next tile's loads fly during compute
            int jt2; int s2 = tile_seg(g + 1, jt2);
            issue_tile(pack.s[s2], jt2, &sh_gb[p ^ 1][0], &sh_ft[p ^ 1][0], tid);
            __builtin_prefetch(pack.s[s2].adj + i * pack.s[s2].adj_stride + jt2, 0, 0);
        }

        // softmax weights for this tile (4 rows x 32 cols = 128 workers)
        if (tid < 4 * TJ) {
            const int ilw = tid >> 5;
            const int jj  = tid & (TJ - 1);
            const int iw  = blockIdx.x * 4 + ilw;
            float w = 0.f;
            if (sg.adj[iw * sg.adj_stride + jt + jj] != 0) {
                float x = sg.pa[iw] + sg.pb[jt + jj];
                float s = (x > 0.f) ? x : (__expf(x) - 1.f);
                w = __expf(s - sg.m[iw]) / sg.z[iw];
            }
            sh_w[p][ilw * TJ + jj] = w;
        }
        __syncthreads();

#pragma unroll
        for (int jj = 0; jj < TJ; ++jj) {
            const float w = sh_w[p][il * TJ + jj];
            const float x = ga + sh_gb[p][jj * DH + d];
            const float gt = (x > 0.f) ? x : (__expf(x) - 1.f);
            acc += w * gt * sh_ft[p][jj * DH + d];
        }
    }
    out[i * DH + d] = acc;
}

// ---------------------------------------------------------------------------
// Host-side launcher
// ---------------------------------------------------------------------------
extern "C" void kernel_launch(void* const* d_in, const int* in_sizes, int n_in,
                              void* d_out, int out_size, void* d_ws, size_t ws_size,
                              hipStream_t stream) {
    (void)in_sizes; (void)n_in; (void)out_size; (void)ws_size;

    const float* sf      = (const float*)d_in[0];
    const float* tf      = (const float*)d_in[1];
    const float* af      = (const float*)d_in[2];
    const float* attr_W  = (const float*)d_in[3];   // (3,64,64)
    const float* attr_b  = (const float*)d_in[4];   // (3,64)
    const float* alpha_W = (const float*)d_in[5];   // (14,64)
    const float* alpha_b = (const float*)d_in[6];   // (14,)
    const int* adj_full  = (const int*)d_in[7];     // (1280,1280)
    const int* adj_sa    = (const int*)d_in[8];     // (1024,256)
    const int* adj_as    = (const int*)d_in[9];     // (256,1024)
    const int* adj_aa    = (const int*)d_in[10];    // (256,256)
    float* out = (float*)d_out;

    // workspace layout (floats): ~508 KB total
    float* ws    = (float*)d_ws;
    float* t1    = ws;                 // NS*DH
    float* t2    = t1 + NS * DH;       // NT*DH
    float* t3    = t2 + NT * DH;       // NA*DH
    float* proj  = t3 + NA * DH;       // 14 * NS (each padded to NS)
    float* stats = proj + 14 * NS;     // 7 pairs * 2 * NS
    const int NFULL = NS + NT;         // 1280

    auto M = [&](int pair) { return stats + pair * 2 * NS; };
    auto Z = [&](int pair) { return stats + pair * 2 * NS + NS; };
    auto P = [&](int k)    { return proj + k * NS; };

    // 1) attribute GEMMs (WMMA)
    attr_gemm_wmma<<<NS / 32, 256, 0, stream>>>(sf, attr_W,               attr_b,          t1);
    attr_gemm_wmma<<<NT / 32, 256, 0, stream>>>(tf, attr_W + 1 * DH * DH, attr_b + DH,     t2);
    attr_gemm_wmma<<<NA / 32, 256, 0, stream>>>(af, attr_W + 2 * DH * DH, attr_b + 2 * DH, t3);

    // 2) alpha projections
    proj_kernel<<<dim3(NS / 128, 14), 128, 0, stream>>>(sf, tf, af, alpha_W, alpha_b, proj);

    // 3) per-row softmax stats for the 7 attention pairs
    rowstats_kernel<<<NS, 256, 0, stream>>>(P(0),  P(1),  adj_full,                   NFULL, NS, M(0), Z(0)); // ss
    rowstats_kernel<<<NS, 256, 0, stream>>>(P(2),  P(3),  adj_full + NS,              NFULL, NT, M(1), Z(1)); // st
    rowstats_kernel<<<NT, 256, 0, stream>>>(P(4),  P(5),  adj_full + NS * NFULL,      NFULL, NS, M(2), Z(2)); // ts
    rowstats_kernel<<<NT, 256, 0, stream>>>(P(6),  P(7),  adj_full + NS * NFULL + NS, NFULL, NT, M(3), Z(3)); // tt
    rowstats_kernel<<<NS, 256, 0, stream>>>(P(8),  P(9),  adj_sa,                     NA,    NA, M(4), Z(4)); // sa
    rowstats_kernel<<<NA, 256, 0, stream>>>(P(10), P(11), adj_as,                     NS,    NS, M(5), Z(5)); // as
    rowstats_kernel<<<NA, 256, 0, stream>>>(P(12), P(13), adj_aa,                     NA,    NA, M(6), Z(6)); // aa

    // 4) fused gated aggregation
    // stock_out = agg(ss) + agg(st) + agg(sa) + sf
    {
        SegPack pk{};
        pk.nseg = 3;
        pk.s[0] = Seg{t1, sf, P(1), P(0), M(0), Z(0), adj_full,                   NFULL, NS};
        pk.s[1] = Seg{t2, tf, P(3), P(2), M(1), Z(1), adj_full + NS,              NFULL, NT};
        pk.s[2] = Seg{t3, af, P(9), P(8), M(4), Z(4), adj_sa,                     NA,    NA};
        agg_kernel<<<NS / 4, 256, 0, stream>>>(pk, t1, sf, out);
    }
    // topic_out = agg(tt) + agg(ts)
    {
        SegPack pk{};
        pk.nseg = 2;
        pk.s[0] = Seg{t2, tf, P(7), P(6), M(3), Z(3), adj_full + NS * NFULL + NS, NFULL, NT};
        pk.s[1] = Seg{t1, sf, P(5), P(4), M(2), Z(2), adj_full + NS * NFULL,      NFULL, NS};
        agg_kernel<<<NT / 4, 256, 0, stream>>>(pk, t2, nullptr, out + NS * DH);
    }
    // analyst_out = agg(aa) + agg(as)
    {
        SegPack pk{};
        pk.nseg = 2;
        pk.s[0] = Seg{t3, af, P(13), P(12), M(6), Z(6), adj_aa,                   NA,    NA};
        pk.s[1] = Seg{t1, sf, P(11), P(10), M(5), Z(5), adj_as,                   NS,    NS};
        agg_kernel<<<NA / 4, 256, 0, stream>>>(pk, t3, nullptr, out + (NS + NT) * DH);
    }
}